// Gate_89163521065187
// MI455X (gfx1250) — compile-verified
//
#include <hip/hip_runtime.h>

typedef float v8f __attribute__((ext_vector_type(8)));
typedef float v2f __attribute__((ext_vector_type(2)));

#define WAVES_PER_BLOCK 4
#define NODES_PER_WAVE 16
#define NODES_PER_BLOCK (WAVES_PER_BLOCK * NODES_PER_WAVE)
#define ROW_STRIDE 132   // 128 floats + 4 pad -> conflict-free strided ds_load_b64

__global__ void zero_out_kernel(float* __restrict__ out, int n) {
    int i = blockIdx.x * blockDim.x + threadIdx.x;
    if (i < n) out[i] = 0.0f;
}

__global__ __launch_bounds__(128)
void gate_scatter_kernel(const float* __restrict__ x,
                         const float* __restrict__ ref,
                         const float* __restrict__ W,
                         const float* __restrict__ bptr,
                         const int*   __restrict__ index,
                         float* __restrict__ out,
                         int N) {
    __shared__ float sW[256];
    __shared__ float tile[WAVES_PER_BLOCK][NODES_PER_WAVE * ROW_STRIDE];
    __shared__ float wbuf[WAVES_PER_BLOCK][16];

    const int tid  = threadIdx.x;
    const int wave = tid >> 5;
    const int lane = tid & 31;

    // Stage gate weight vector (256 floats) once per block.
    sW[tid]       = W[tid];
    sW[tid + 128] = W[tid + 128];
    __syncthreads();

    const int base = blockIdx.x * NODES_PER_BLOCK + wave * NODES_PER_WAVE;
    int nvalid = N - base;
    if (nvalid < 0)  nvalid = 0;
    if (nvalid > 16) nvalid = 16;

    const int row = lane & 15;          // A-matrix row (node within tile)
    const int hi2 = (lane >> 4) << 1;   // K sub-offset: 0 for lanes 0-15, 2 for lanes 16-31
    float* t = tile[wave];

    v8f c = {};   // 16x16 f32 accumulator (all columns identical by construction)

    // ---------- phase 1: x features (K = 0..127) ----------
    for (int r = 0; r < 16; ++r) {
        float4 v = make_float4(0.f, 0.f, 0.f, 0.f);
        if (r < nvalid)
            v = *(const float4*)(x + (size_t)(base + r) * 128 + lane * 4);
        *(float4*)(t + r * ROW_STRIDE + lane * 4) = v;   // coalesced -> LDS
    }
    for (int k = 0; k < 32; ++k) {
        const int kk = k * 4 + hi2;
        v2f a  = *(const v2f*)(t + row * ROW_STRIDE + kk);  // A fragment
        v2f bw = *(const v2f*)(sW + kk);                    // B = W broadcast over N
        c = __builtin_amdgcn_wmma_f32_16x16x4_f32(false, a, false, bw,
                                                  (short)0, c, false, false);
    }

    // ---------- phase 2: ref features (K = 128..255), reuse same LDS tile ----------
    for (int r = 0; r < 16; ++r) {
        float4 v = make_float4(0.f, 0.f, 0.f, 0.f);
        if (r < nvalid)
            v = *(const float4*)(ref + (size_t)(base + r) * 128 + lane * 4);
        *(float4*)(t + r * ROW_STRIDE + lane * 4) = v;
    }
    for (int k = 0; k < 32; ++k) {
        const int kk = k * 4 + hi2;
        v2f a  = *(const v2f*)(t + row * ROW_STRIDE + kk);
        v2f bw = *(const v2f*)(sW + 128 + kk);
        c = __builtin_amdgcn_wmma_f32_16x16x4_f32(false, a, false, bw,
                                                  (short)0, c, false, false);
    }

    // Gate: w = tanh(dot + bias). D layout: lanes 0-15 hold nodes 0..7 in c[0..7],
    // lanes 16-31 hold nodes 8..15. Publish via LDS (wave-coherent, DS in-order).
    const float bias = bptr[0];
    if (lane == 0) {
        #pragma unroll
        for (int r = 0; r < 8; ++r) wbuf[wave][r] = tanhf(c[r] + bias);
    }
    if (lane == 16) {
        #pragma unroll
        for (int r = 0; r < 8; ++r) wbuf[wave][8 + r] = tanhf(c[r] + bias);
    }

    // Scatter: out[idx] += w * x_row. x row re-read coalesced (L2-resident).
    for (int m = 0; m < nvalid; ++m) {
        const float wm  = wbuf[wave][m];
        const int node  = base + m;
        const int idx   = index[node];                     // wave-uniform -> SMEM
        const float4 xv = *(const float4*)(x + (size_t)node * 128 + lane * 4);
        float* o = out + (size_t)idx * 128 + lane * 4;
        unsafeAtomicAdd(o + 0, wm * xv.x);
        unsafeAtomicAdd(o + 1, wm * xv.y);
        unsafeAtomicAdd(o + 2, wm * xv.z);
        unsafeAtomicAdd(o + 3, wm * xv.w);
    }
}

extern "C" void kernel_launch(void* const* d_in, const int* in_sizes, int n_in,
                              void* d_out, int out_size, void* d_ws, size_t ws_size,
                              hipStream_t stream) {
    const float* x     = (const float*)d_in[0];   // [N,128]
    const float* ref   = (const float*)d_in[1];   // [N,128]
    const float* W     = (const float*)d_in[2];   // [1,256]
    const float* b     = (const float*)d_in[3];   // [1]
    const int*   index = (const int*)d_in[4];     // [N]
    float* out = (float*)d_out;                   // [B,128]

    const int N = in_sizes[4];

    // Output is poisoned by harness -> zero it first (same stream, serialized).
    zero_out_kernel<<<(out_size + 255) / 256, 256, 0, stream>>>(out, out_size);

    const int blocks = (N + NODES_PER_BLOCK - 1) / NODES_PER_BLOCK;
    gate_scatter_kernel<<<blocks, 128, 0, stream>>>(x, ref, W, b, index, out, N);
}